// DotInteraction_10874857194076
// MI455X (gfx1250) — compile-verified
//
#include <hip/hip_runtime.h>
#include <hip/hip_bf16.h>

// fp32 WMMA fragment types
typedef float v2f __attribute__((ext_vector_type(2)));
typedef float v8f __attribute__((ext_vector_type(8)));

#define NUM_FEATURES  64
#define FEATURE_DIM   128
#define LDS_STRIDE    132   // 128 + 4 pad floats: bank = (4*row + k) % 64 -> conflict-free b64 reads
#define OUT_PER_BATCH 2016  // 64*63/2
#define NWAVES        10    // one wave per 16x16 tile of the 4x4 lower-triangle tile grid
#define BLOCK_THREADS (NWAVES * 32)

__global__ __launch_bounds__(BLOCK_THREADS)
void DotInteraction_wmma_f32_kernel(const float* __restrict__ x,
                                    float* __restrict__ out) {
  __shared__ float lds[NUM_FEATURES * LDS_STRIDE];

  const int b   = blockIdx.x;
  const int tid = threadIdx.x;
  const float* xb = x + (size_t)b * (NUM_FEATURES * FEATURE_DIM);

  // ---- Stage X_b (64 x 128 f32 = 32 KB) into LDS with padded rows ----
  // 2048 float4 elements, fully coalesced; row stride 132 floats (16B aligned).
  for (int idx = tid; idx < (NUM_FEATURES * FEATURE_DIM) / 4; idx += BLOCK_THREADS) {
    const int row = idx >> 5;   // 32 float4 per source row
    const int c4  = idx & 31;
    float4 v = ((const float4*)xb)[idx];
    *(float4*)(&lds[row * LDS_STRIDE + c4 * 4]) = v;
  }
  __syncthreads();

  const int wave = tid >> 5;   // wave32
  const int lane = tid & 31;

  // wave -> (ti, tj), ti >= tj, row-major over the lower triangle incl. diagonal:
  // (0,0),(1,0),(1,1),(2,0),(2,1),(2,2),(3,0),(3,1),(3,2),(3,3)
  const int ti = (wave >= 6) ? 3 : (wave >= 3) ? 2 : (wave >= 1) ? 1 : 0;
  const int tj = wave - (ti * (ti + 1)) / 2;

  // Fragment addressing (identical maps for A and B of a Gram matrix):
  //   A 16x4: lane l -> row l&15,  K = 2*(l>>4) + v
  //   B 4x16: lane l -> col l&15,  K = v + 2*(l>>4)
  const int lrow = lane & 15;
  const int half = lane >> 4;
  const float* arow = &lds[(ti * 16 + lrow) * LDS_STRIDE + half * 2];
  const float* brow = &lds[(tj * 16 + lrow) * LDS_STRIDE + half * 2];

  v8f c = {0.f, 0.f, 0.f, 0.f, 0.f, 0.f, 0.f, 0.f};
  #pragma unroll
  for (int k = 0; k < FEATURE_DIM; k += 4) {
    v2f a  = *(const v2f*)(arow + k);
    v2f bb = *(const v2f*)(brow + k);
    // 8 args: (neg_a, A, neg_b, B, c_mod, C, reuse_a, reuse_b)
    c = __builtin_amdgcn_wmma_f32_16x16x4_f32(
        /*neg_a=*/false, a, /*neg_b=*/false, bb,
        /*c_mod=*/(short)0, c, /*reuse_a=*/false, /*reuse_b=*/false);
  }

  // ---- Scatter strict lower triangle: out[b][i*(i-1)/2 + j] for i > j ----
  // C/D layout: lane l -> N = l&15; VGPR r -> M = r + 8*(l>>4)
  float* outb = out + (size_t)b * OUT_PER_BATCH;
  const int jj    = tj * 16 + (lane & 15);
  const int ibase = ti * 16 + (lane >> 4) * 8;
  #pragma unroll
  for (int r = 0; r < 8; ++r) {
    const int ii = ibase + r;
    if (ii > jj) {
      outb[(ii * (ii - 1)) / 2 + jj] = c[r];
    }
  }
}

extern "C" void kernel_launch(void* const* d_in, const int* in_sizes, int n_in,
                              void* d_out, int out_size, void* d_ws, size_t ws_size,
                              hipStream_t stream) {
  (void)n_in; (void)d_ws; (void)ws_size; (void)out_size;
  const float* x  = (const float*)d_in[0];
  float* out      = (float*)d_out;
  const int batch = in_sizes[0] / (NUM_FEATURES * FEATURE_DIM);  // 8192
  DotInteraction_wmma_f32_kernel<<<batch, BLOCK_THREADS, 0, stream>>>(x, out);
}